// SNNLoss_7756710936619
// MI455X (gfx1250) — compile-verified
//
#include <hip/hip_runtime.h>
#include <math.h>

// SNN loss, fused WMMA implementation for gfx1250 (MI455X).
// x: [8192,256] f32, y: [8192] i64, T: [1] f32 -> scalar f32 loss.

typedef float v2f __attribute__((ext_vector_type(2)));
typedef float v8f __attribute__((ext_vector_type(8)));
typedef int   gv4i __attribute__((vector_size(16)));   // matches builtin param type

#define NROWS 8192
#define FEAT  256
#define AS    260   // LDS stride for A tile: 4-float aligned, bank-stride 4
#define BS    68    // LDS stride for B tile: 4-float aligned, bank-stride 4
#define NCHUNK (NROWS / 128 * 4)   // 256 B-chunks of 128 cols x 64 K

#if __has_builtin(__builtin_amdgcn_global_load_async_to_lds_b128)
#define HAS_ASYNC 1
#define ASYNC_B128(g, l)                                                 \
    __builtin_amdgcn_global_load_async_to_lds_b128(                      \
        (__attribute__((address_space(1))) gv4i*)(g),                    \
        (__attribute__((address_space(3))) gv4i*)(l), 0, 0)
#else
#define HAS_ASYNC 0
#endif

#if __has_builtin(__builtin_amdgcn_s_wait_asynccnt)
#define WAIT_ASYNC(n) __builtin_amdgcn_s_wait_asynccnt(n)
#else
#define WAIT_ASYNC(n) asm volatile("s_wait_asynccnt %0" ::"i"(n))
#endif

__device__ __forceinline__ void lse_add(float& m, float& s, float v) {
    if (v > m) { s = s * __expf(m - v) + 1.0f; m = v; }
    else       { s += __expf(v - m); }
}

__device__ __forceinline__ void lse_merge(float& m, float& s, float m2, float s2) {
    if (s2 == 0.0f) return;
    if (s == 0.0f) { m = m2; s = s2; return; }
    if (m2 > m) { s = s * __expf(m - m2) + s2; m = m2; }
    else        { s += s2 * __expf(m2 - m); }
}

// 16 x v_wmma_f32_16x16x4_f32 over a 64-wide K chunk
__device__ __forceinline__ void wmma_chunk(const float* __restrict__ lAp,
                                           const float* __restrict__ lBp,
                                           int lo, int wave, int koff, int kc,
                                           v8f& c) {
    const float* aBase = &lAp[lo * AS + kc * 64 + koff];
    const float* bBase = &lBp[(wave * 16 + lo) * BS + koff];
#pragma unroll
    for (int k = 0; k < 64; k += 4) {
        v2f a = *(const v2f*)(aBase + k);
        v2f b = *(const v2f*)(bBase + k);
        c = __builtin_amdgcn_wmma_f32_16x16x4_f32(
                false, a, false, b, (short)0, c, false, false);
    }
}

// ---- Kernel 1: per-row sum & sum-of-squares (one wave per row) -------------
__global__ void __launch_bounds__(256) snn_rowstats(const float* __restrict__ x,
                                                    float* __restrict__ sq,
                                                    float* __restrict__ rowsum) {
    const int wave = threadIdx.x >> 5;
    const int lane = threadIdx.x & 31;
    const int row  = blockIdx.x * 8 + wave;
    const float* xr = x + row * FEAT;
    float s = 0.0f, q = 0.0f;
#pragma unroll
    for (int i = 0; i < FEAT; i += 32) {
        float v = xr[i + lane];
        s += v;
        q += v * v;
    }
#pragma unroll
    for (int m = 16; m >= 1; m >>= 1) {
        s += __shfl_xor(s, m, 32);
        q += __shfl_xor(q, m, 32);
    }
    if (lane == 0) { sq[row] = q; rowsum[row] = s; }
}

// ---- Kernel 2: deterministic global std (ddof=1) -> scale = 10^T / std ----
__global__ void __launch_bounds__(256) snn_scale(const float* __restrict__ sq,
                                                 const float* __restrict__ rowsum,
                                                 const float* __restrict__ T,
                                                 float* __restrict__ scale_out) {
    __shared__ float redS[256];
    __shared__ float redQ[256];
    float s = 0.0f, q = 0.0f;
    for (int i = threadIdx.x; i < NROWS; i += 256) { s += rowsum[i]; q += sq[i]; }
    redS[threadIdx.x] = s;
    redQ[threadIdx.x] = q;
    __syncthreads();
    for (int stride = 128; stride >= 1; stride >>= 1) {
        if (threadIdx.x < stride) {
            redS[threadIdx.x] += redS[threadIdx.x + stride];
            redQ[threadIdx.x] += redQ[threadIdx.x + stride];
        }
        __syncthreads();
    }
    if (threadIdx.x == 0) {
        double n   = (double)NROWS * (double)FEAT;
        double S   = (double)redS[0];
        double Q   = (double)redQ[0];
        double var = (Q - S * S / n) / (n - 1.0);
        scale_out[0] = exp10f(T[0]) / (float)sqrt(var);
    }
}

// ---- Kernel 3: fused Gram(WMMA f32 16x16x4) + online masked logsumexp -----
// Block: 256 threads (8 waves). Block owns 16 rows; waves tile 128 columns.
__global__ void __launch_bounds__(256) snn_main(const float* __restrict__ x,
                                                const long long* __restrict__ y,
                                                const float* __restrict__ sq,
                                                const float* __restrict__ scale_p,
                                                float* __restrict__ terms) {
    __shared__ float lA[16 * AS];
#if HAS_ASYNC
    __shared__ float lB[2][128 * BS];   // double-buffered for async pipeline
#else
    __shared__ float lB[128 * BS];
#endif
    __shared__ float red[8][16][4];

    const int tid   = threadIdx.x;
    const int wave  = tid >> 5;
    const int lane  = tid & 31;
    const int lo    = lane & 15;
    const int hi    = lane >> 4;
    const int koff  = hi * 2;
    const int rowBase = blockIdx.x * 16;
    const float scale = scale_p[0];

    // Stage A tile (16 rows x 256) in LDS, vectorized: 4 float4 per thread.
#pragma unroll
    for (int it = 0; it < 4; ++it) {
        int idx = tid + it * 256;       // float4 index, 0..1023
        int r   = idx >> 6;             // 64 float4 per row
        int k4  = idx & 63;
        *(float4*)&lA[r * AS + k4 * 4] =
            *(const float4*)&x[(rowBase + r) * FEAT + k4 * 4];
    }

    // Per-lane row metadata.
    float sqrow[8];
    int   yrow[8];
#pragma unroll
    for (int v = 0; v < 8; ++v) {
        int r = rowBase + v + hi * 8;
        sqrow[v] = sq[r];
        yrow[v]  = (int)y[r];
    }

    float mden[8], sden[8], mnum[8], snum[8];
#pragma unroll
    for (int v = 0; v < 8; ++v) {
        mden[v] = -INFINITY; sden[v] = 0.0f;
        mnum[v] = -INFINITY; snum[v] = 0.0f;
    }

#if HAS_ASYNC
    // ---------- async global->LDS double-buffered pipeline ----------
    // chunk t: colBase=(t>>2)*128, kc=t&3 ; buffer t&1
    {
#pragma unroll
        for (int it = 0; it < 8; ++it) {
            int idx = tid + it * 256;   // float4 index, 0..2047
            int col = idx >> 4;         // 16 float4 per column chunk
            int k4  = idx & 15;
            ASYNC_B128(&x[col * FEAT + k4 * 4], &lB[0][col * BS + k4 * 4]);
        }
    }
    int t = 0;
    for (int jt = 0; jt < NROWS / 128; ++jt) {
        v8f c = {};
#pragma unroll
        for (int kc = 0; kc < 4; ++kc) {
            const int nt = t + 1;
            if (nt < NCHUNK) {
                const int ncol = (nt >> 2) * 128;
                const int nkc  = nt & 3;
                float* buf = lB[nt & 1];
#pragma unroll
                for (int it = 0; it < 8; ++it) {
                    int idx = tid + it * 256;
                    int col = idx >> 4;
                    int k4  = idx & 15;
                    ASYNC_B128(&x[(ncol + col) * FEAT + nkc * 64 + k4 * 4],
                               &buf[col * BS + k4 * 4]);
                }
                WAIT_ASYNC(8);          // chunk t landed; chunk t+1 in flight
            } else {
                WAIT_ASYNC(0);
            }
            __syncthreads();
            wmma_chunk(lA, lB[t & 1], lo, wave, koff, kc, c);
            __syncthreads();
            ++t;
        }
#else
    // ---------- fallback: register-staged pipeline ----------
    float4 rcur[8], rnext[8];
#pragma unroll
    for (int it = 0; it < 8; ++it) {
        int idx = tid + it * 256;
        int col = idx >> 4;
        int k4  = idx & 15;
        rcur[it] = *(const float4*)&x[col * FEAT + k4 * 4];
    }
    int t = 0;
    for (int jt = 0; jt < NROWS / 128; ++jt) {
        v8f c = {};
#pragma unroll
        for (int kc = 0; kc < 4; ++kc) {
            const int nt = t + 1;
            __syncthreads();            // chunk t-1 consumers done with lB
#pragma unroll
            for (int it = 0; it < 8; ++it) {
                int idx = tid + it * 256;
                int col = idx >> 4;
                int k4  = idx & 15;
                *(float4*)&lB[col * BS + k4 * 4] = rcur[it];
            }
            if (nt < NCHUNK) {
                const int ncol = (nt >> 2) * 128;
                const int nkc  = nt & 3;
#pragma unroll
                for (int it = 0; it < 8; ++it) {
                    int idx = tid + it * 256;
                    int col = idx >> 4;
                    int k4  = idx & 15;
                    rnext[it] = *(const float4*)&x[(ncol + col) * FEAT + nkc * 64 + k4 * 4];
                }
            }
            __syncthreads();
            wmma_chunk(lA, lB, lo, wave, koff, kc, c);
#pragma unroll
            for (int it = 0; it < 8; ++it) rcur[it] = rnext[it];
            ++t;
        }
#endif
        // Epilogue for this 16x128 stripe of the Gram matrix.
        const int colBase = jt * 128;
        const int col  = colBase + wave * 16 + lo;
        const float scol = sq[col];
        const int  ycol  = (int)y[col];
#pragma unroll
        for (int v = 0; v < 8; ++v) {
            int row = rowBase + v + hi * 8;
            if (row != col) {
                float d2 = fmaxf(sqrow[v] + scol - 2.0f * c[v], 0.0f);
                float e  = -sqrtf(d2) * scale;
                lse_add(mden[v], sden[v], e);
                if (yrow[v] == ycol) lse_add(mnum[v], snum[v], e);
            }
        }
    }

    // Combine the 16 column-lanes of each half-wave.
#pragma unroll
    for (int v = 0; v < 8; ++v) {
#pragma unroll
        for (int msk = 1; msk <= 8; msk <<= 1) {
            float m2 = __shfl_xor(mden[v], msk, 32);
            float s2 = __shfl_xor(sden[v], msk, 32);
            lse_merge(mden[v], sden[v], m2, s2);
            m2 = __shfl_xor(mnum[v], msk, 32);
            s2 = __shfl_xor(snum[v], msk, 32);
            lse_merge(mnum[v], snum[v], m2, s2);
        }
    }
    if (lo == 0) {
#pragma unroll
        for (int v = 0; v < 8; ++v) {
            int r = v + hi * 8;
            red[wave][r][0] = mden[v];
            red[wave][r][1] = sden[v];
            red[wave][r][2] = mnum[v];
            red[wave][r][3] = snum[v];
        }
    }
    __syncthreads();

    if (tid < 16) {
        float md = -INFINITY, sd = 0.0f, mn = -INFINITY, sn = 0.0f;
#pragma unroll
        for (int w = 0; w < 8; ++w) {
            lse_merge(md, sd, red[w][tid][0], red[w][tid][1]);
            lse_merge(mn, sn, red[w][tid][2], red[w][tid][3]);
        }
        float den  = __logf(sd) + md;
        float term = (sn > 0.0f) ? (__logf(sn) + mn - den) : 0.0f;
        terms[rowBase + tid] = term;
    }
}

// ---- Kernel 4: deterministic final mean ------------------------------------
__global__ void __launch_bounds__(256) snn_finalize(const float* __restrict__ terms,
                                                    float* __restrict__ out) {
    __shared__ float red[256];
    float s = 0.0f;
    for (int i = threadIdx.x; i < NROWS; i += 256) s += terms[i];
    red[threadIdx.x] = s;
    __syncthreads();
    for (int stride = 128; stride >= 1; stride >>= 1) {
        if (threadIdx.x < stride) red[threadIdx.x] += red[threadIdx.x + stride];
        __syncthreads();
    }
    if (threadIdx.x == 0) out[0] = -red[0] / (float)NROWS;
}

extern "C" void kernel_launch(void* const* d_in, const int* in_sizes, int n_in,
                              void* d_out, int out_size, void* d_ws, size_t ws_size,
                              hipStream_t stream) {
    const float*     x = (const float*)d_in[0];
    const long long* y = (const long long*)d_in[1];
    const float*     T = (const float*)d_in[2];
    float* ws     = (float*)d_ws;
    float* scale  = ws;                       // [0]
    float* sq     = ws + 64;                  // [8192]
    float* rowsum = sq + NROWS;               // [8192]
    float* terms  = rowsum + NROWS;           // [8192]
    float* out    = (float*)d_out;

    snn_rowstats<<<NROWS / 8, 256, 0, stream>>>(x, sq, rowsum);
    snn_scale<<<1, 256, 0, stream>>>(sq, rowsum, T, scale);
    snn_main<<<NROWS / 16, 256, 0, stream>>>(x, y, sq, scale, terms);
    snn_finalize<<<1, 256, 0, stream>>>(terms, out);
}